// AttentionLayer_77017353552603
// MI455X (gfx1250) — compile-verified
//
#include <hip/hip_runtime.h>
#include <stdint.h>

// Problem constants (match reference)
#define Bv     32
#define Sv     4096
#define Dv     1024
#define NEGF   (-1e37f)

// Tuning
#define CHUNK  256                 // S rows per workgroup (== blockDim)
#define NCHUNK (Sv / CHUNK)        // 16 -> 512 workgroups in kernel 1
#define TPB    256
#define WAVES  (TPB / 32)
#define TPAD   33                  // padded LDS row stride (floats) for 16x32 tile

typedef __attribute__((ext_vector_type(2))) float v2f;
typedef __attribute__((ext_vector_type(8))) float v8f;

// ---- CDNA5 helpers -------------------------------------------------------

// Byte offset of a __shared__ object inside the workgroup's LDS block.
// flat -> local addrspacecast lowers to truncation of the flat address.
__device__ __forceinline__ uint32_t lds_byte_off(const void* p) {
  return (uint32_t)(uintptr_t)(__attribute__((address_space(3))) const void*)p;
}

// GLOBAL_LOAD_ASYNC_TO_LDS_B128: copies 16 bytes per lane from global memory
// directly into LDS (LDS dest address comes from a VGPR). Tracked by ASYNCcnt.
__device__ __forceinline__ void async_copy_b128(uint32_t lds_off, const float* gaddr) {
  asm volatile("global_load_async_to_lds_b128 %0, %1, off"
               :: "v"(lds_off), "v"(gaddr)
               : "memory");
}

__device__ __forceinline__ void wait_async0() {
  asm volatile("s_wait_asynccnt 0" ::: "memory");
}

// --------------------------------------------------------------------------
// Kernel 1: flash-decode partials. Each workgroup handles one (batch, chunk)
// of 256 S-rows: computes masked logits via WMMA, a chunk-local softmax
// (max m, sum l), and the unnormalized weighted V-sum o[D].
// --------------------------------------------------------------------------
__global__ __launch_bounds__(TPB)
void attn_partial_kernel(const float* __restrict__ Kp,
                         const float* __restrict__ Qp,
                         const float* __restrict__ Vp,
                         const int*   __restrict__ Mp,
                         float* __restrict__ ws)
{
  __shared__ float q_lds[Dv];                  // 4 KB: query row, read-broadcast
  __shared__ float tile[WAVES][16 * TPAD];     // ~16.5 KB: per-wave K tiles
  __shared__ float Lp[CHUNK];                  // logits, then softmax weights
  __shared__ float redm[WAVES];
  __shared__ float reds[WAVES];

  const int tid  = threadIdx.x;
  const int wave = tid >> 5;
  const int lane = tid & 31;
  const int half = lane >> 4;      // which 16-lane half of the wave
  const int mrow = lane & 15;

  const int b      = blockIdx.x / NCHUNK;
  const int c      = blockIdx.x % NCHUNK;
  const int s_base = c * CHUNK;    // chunk's first S row

  // Preload the query row once per workgroup.
  for (int i = tid; i < Dv; i += TPB) q_lds[i] = Qp[(size_t)b * Dv + i];
  __syncthreads();

  const float* Kb = Kp + (size_t)b * Sv * Dv;

  // ---- Phase A: logits via V_WMMA_F32_16X16X4_F32 ----
  // Wave w owns rows [w*32, w*32+32): two 16-row WMMA tiles.
  // A[m][k] = K[s0+m][d0+kk+k]  (staged in LDS),  B[k][n] = q[d0+kk+k] (bcast
  // over N), so D[m][n] accumulates logit(s0+m) in every column n.
  for (int t = 0; t < 2; ++t) {
    const int s0 = wave * 32 + t * 16;                 // local row base
    v8f acc = {0.f, 0.f, 0.f, 0.f, 0.f, 0.f, 0.f, 0.f};
    const uint32_t tbase = lds_byte_off(&tile[wave][0]);

    for (int d0 = 0; d0 < Dv; d0 += 32) {
      // Stage 16x32 fp32 K-tile: 4x b128 async copies per wave, coalesced
      // along rows in global memory, row-padded in LDS.
#pragma unroll
      for (int it = 0; it < 4; ++it) {
        const int f   = it * 128 + lane * 4;           // flat float index
        const int row = f >> 5;
        const int col = f & 31;
        const float* gp = Kb + (size_t)(s_base + s0 + row) * Dv + d0 + col;
        async_copy_b128(tbase + (uint32_t)((row * TPAD + col) * 4), gp);
      }
      wait_async0();

#pragma unroll
      for (int kk = 0; kk < 32; kk += 4) {
        v2f a, bb;
        // 32-bit A 16x4 layout: lanes 0-15 hold K=0,1; lanes 16-31 hold K=2,3
        a.x  = tile[wave][mrow * TPAD + kk + 2 * half];
        a.y  = tile[wave][mrow * TPAD + kk + 2 * half + 1];
        bb.x = q_lds[d0 + kk + 2 * half];
        bb.y = q_lds[d0 + kk + 2 * half + 1];
        acc = __builtin_amdgcn_wmma_f32_16x16x4_f32(
            /*neg_a=*/false, a, /*neg_b=*/false, bb,
            /*c_mod=*/(short)0, acc, /*reuse_a=*/false, /*reuse_b=*/false);
      }
    }

    // Extract: C/D layout -> lane 0 holds rows 0..7 in VGPRs 0..7,
    // lane 16 holds rows 8..15. Apply mask while writing.
    if (mrow == 0) {
#pragma unroll
      for (int i = 0; i < 8; ++i) {
        const int r  = half * 8 + i;
        const int sg = s_base + s0 + r;
        float lv = acc[i];
        if (Mp[(size_t)b * Sv + sg] == 0) lv = NEGF;
        Lp[s0 + r] = lv;
      }
    }
  }
  __syncthreads();

  // ---- Phase B: chunk-local softmax (block max + sum) ----
  const float v = Lp[tid];
  float mx = v;
#pragma unroll
  for (int o = 16; o > 0; o >>= 1) mx = fmaxf(mx, __shfl_xor(mx, o, 32));
  if (lane == 0) redm[wave] = mx;
  __syncthreads();
  float m = redm[0];
#pragma unroll
  for (int i = 1; i < WAVES; ++i) m = fmaxf(m, redm[i]);

  const float p = __expf(v - m);   // masked rows underflow to exactly 0
  Lp[tid] = p;
  float ps = p;
#pragma unroll
  for (int o = 16; o > 0; o >>= 1) ps += __shfl_xor(ps, o, 32);
  if (lane == 0) reds[wave] = ps;
  __syncthreads();
  float l = 0.f;
#pragma unroll
  for (int i = 0; i < WAVES; ++i) l += reds[i];

  // ---- Phase C: o[d] = sum_s p[s] * V[s][d], skipping zero-weight rows ----
  float a0 = 0.f, a1 = 0.f, a2 = 0.f, a3 = 0.f;
  const float* Vb = Vp + ((size_t)b * Sv + s_base) * Dv;
  for (int s = 0; s < CHUNK; ++s) {
    const float w = Lp[s];                 // uniform across the block
    if (w != 0.f) {                        // coherent branch: skips masked rows
      const float* vr = Vb + (size_t)s * Dv + tid;
      a0 += w * vr[0];
      a1 += w * vr[256];
      a2 += w * vr[512];
      a3 += w * vr[768];
    }
  }

  // ---- Write partials ----
  const int pid = blockIdx.x;              // == b*NCHUNK + c
  float* mls = ws;                         // [B*NCHUNK][2] : m, l
  float* os  = ws + (size_t)Bv * NCHUNK * 2;
  if (tid == 0) { mls[pid * 2] = m; mls[pid * 2 + 1] = l; }
  float* op = os + (size_t)pid * Dv;
  op[tid]       = a0;
  op[tid + 256] = a1;
  op[tid + 512] = a2;
  op[tid + 768] = a3;
}

// --------------------------------------------------------------------------
// Kernel 2: combine the NCHUNK partials per batch with log-sum-exp rescaling.
// --------------------------------------------------------------------------
__global__ __launch_bounds__(TPB)
void attn_combine_kernel(const float* __restrict__ ws, float* __restrict__ out)
{
  const int b   = blockIdx.x;
  const int tid = threadIdx.x;
  const float* mls = ws;
  const float* os  = ws + (size_t)Bv * NCHUNK * 2;

  float mv[NCHUNK], lv[NCHUNK];
#pragma unroll
  for (int i = 0; i < NCHUNK; ++i) {
    mv[i] = mls[(b * NCHUNK + i) * 2];
    lv[i] = mls[(b * NCHUNK + i) * 2 + 1];
  }
  float M = mv[0];
#pragma unroll
  for (int i = 1; i < NCHUNK; ++i) M = fmaxf(M, mv[i]);

  float wgt[NCHUNK];
  float L = 0.f;
#pragma unroll
  for (int i = 0; i < NCHUNK; ++i) {
    wgt[i] = __expf(mv[i] - M);   // all-masked chunks get weight 0
    L += wgt[i] * lv[i];
  }
  const float inv = 1.f / L;

#pragma unroll
  for (int j = 0; j < 4; ++j) {
    const int d = tid + 256 * j;
    float acc = 0.f;
#pragma unroll
    for (int i = 0; i < NCHUNK; ++i)
      acc += wgt[i] * os[(size_t)(b * NCHUNK + i) * Dv + d];
    out[(size_t)b * Dv + d] = acc * inv;
  }
}

// --------------------------------------------------------------------------
extern "C" void kernel_launch(void* const* d_in, const int* in_sizes, int n_in,
                              void* d_out, int out_size, void* d_ws, size_t ws_size,
                              hipStream_t stream) {
  (void)in_sizes; (void)n_in; (void)out_size; (void)ws_size;
  const float* Kp = (const float*)d_in[0];   // keys    [B,S,D] f32
  const float* Qp = (const float*)d_in[1];   // queries [B,1,D] f32
  const float* Vp = (const float*)d_in[2];   // values  [B,S,D] f32
  const int*   Mp = (const int*)d_in[3];     // mask    [B,S]   (nonzero = keep)
  float* out = (float*)d_out;                // [B,1,D] f32
  float* ws  = (float*)d_ws;                 // >= (B*NCHUNK*(2+D))*4 bytes ~= 2.1 MB

  attn_partial_kernel<<<dim3(Bv * NCHUNK), dim3(TPB), 0, stream>>>(Kp, Qp, Vp, Mp, ws);
  attn_combine_kernel<<<dim3(Bv), dim3(TPB), 0, stream>>>(ws, out);
}